// GCL_24833500905739
// MI455X (gfx1250) — compile-verified
//
#include <hip/hip_runtime.h>
#include <math.h>

#define N_NODES 50000
#define N_EDGES 800000
#define HID     128
#define SHS     132   // padded LDS row stride (floats): stride%64==4 -> conflict-free column walks

typedef __attribute__((ext_vector_type(2))) float v2f;
typedef __attribute__((ext_vector_type(8))) float v8f;

// ---------------- kernel 1: zero the aggregate scratch ----------------
__global__ __launch_bounds__(256) void GCL_zero_agg(float* __restrict__ agg) {
    int i = blockIdx.x * 256 + threadIdx.x;
    if (i < N_NODES) agg[i] = 0.0f;
}

// ---------------- kernel 2: scatter-add distances into agg ----------------
__global__ __launch_bounds__(256) void GCL_scatter(const int* __restrict__ edges,
                                                   const float* __restrict__ dist,
                                                   float* __restrict__ agg) {
    int e = blockIdx.x * 256 + threadIdx.x;
    if (e < N_EDGES) {
        int row = edges[e];              // edges[0][e] (row indices are the first E entries)
        unsafeAtomicAdd(&agg[row], dist[e]);  // -> global_atomic_add_f32
    }
}

// ---------------- kernel 3: fused node MLP with residual (fp32 WMMA) ----------------
// out = h + silu([h, agg/100] @ W_n1 + b_n1) @ W_n2 + b_n2
// block: 256 threads = 8 waves; 16 nodes per block; wave w owns output cols [16w,16w+16)
__global__ __launch_bounds__(256) void GCL_node_mlp(const float* __restrict__ h,
                                                    const float* __restrict__ agg,
                                                    const float* __restrict__ Wn1,
                                                    const float* __restrict__ bn1,
                                                    const float* __restrict__ Wn2,
                                                    const float* __restrict__ bn2,
                                                    float* __restrict__ out) {
    __shared__ float sH[16 * SHS];   // h tile   [16 x 128]
    __shared__ float sT[16 * SHS];   // silu(l1) [16 x 128]
    __shared__ float sAgg[16];

    const int tid = threadIdx.x;
    const int m0  = blockIdx.x * 16;              // 50000 = 3125 * 16, no tail

    // ---- stage h tile (16x128 fp32 = 8KB) coalesced as float4 ----
    {
        const float4* hv = (const float4*)(h + (size_t)m0 * HID);
        #pragma unroll
        for (int i = 0; i < 2; ++i) {
            int e = tid + i * 256;                // 512 float4s total
            int r = e >> 5;                       // 32 float4 per row
            int c = e & 31;
            float4 v = hv[r * 32 + c];
            *(float4*)&sH[r * SHS + c * 4] = v;
        }
    }
    if (tid < 16) sAgg[tid] = agg[m0 + tid] * 0.01f;   // fold the /100
    __syncthreads();

    const int lane = tid & 31;
    const int wave = tid >> 5;
    const int n0   = wave * 16;
    const int half = lane >> 4;       // 0: lanes 0-15, 1: lanes 16-31
    const int ln   = lane & 15;
    const int n    = n0 + ln;         // this lane's output column

    // ===== layer 1: acc = bias + agg * W_n1[128,:]  (rank-1 K=129th term), then 32x WMMA K=4 =====
    v8f acc;
    {
        const float w128 = Wn1[128 * HID + n];
        const float bb   = bn1[n];
        #pragma unroll
        for (int r = 0; r < 8; ++r) {
            int m = r + 8 * half;                 // C/D layout: VGPR r -> M = r + 8*half
            acc[r] = bb + sAgg[m] * w128;
        }
    }
    #pragma unroll 8
    for (int k = 0; k < HID; k += 4) {
        v2f a, b;
        const float* ap = &sH[ln * SHS + k + 2 * half];   // A: M=ln, K = k+2*half, k+2*half+1
        a.x = ap[0];
        a.y = ap[1];
        b.x = Wn1[(k + 2 * half) * HID + n];              // B: N=n, same K split
        b.y = Wn1[(k + 2 * half + 1) * HID + n];
        acc = __builtin_amdgcn_wmma_f32_16x16x4_f32(false, a, false, b,
                                                    (short)0, acc, false, false);
    }

    // ---- SiLU (fast path: v_exp_f32 + v_rcp_f32), share across waves ----
    #pragma unroll
    for (int r = 0; r < 8; ++r) {
        float x = acc[r];
        float s = __builtin_amdgcn_rcpf(1.0f + __expf(-x));  // sigmoid, ~1ulp
        int m = r + 8 * half;
        sT[m * SHS + n] = x * s;
    }
    __syncthreads();

    // ===== layer 2: acc2 = h + bias (residual folded into C), then 32x WMMA K=4 =====
    v8f acc2;
    {
        const float bb = bn2[n];
        #pragma unroll
        for (int r = 0; r < 8; ++r) {
            int m = r + 8 * half;
            acc2[r] = bb + sH[m * SHS + n];
        }
    }
    #pragma unroll 8
    for (int k = 0; k < HID; k += 4) {
        v2f a, b;
        const float* ap = &sT[ln * SHS + k + 2 * half];
        a.x = ap[0];
        a.y = ap[1];
        b.x = Wn2[(k + 2 * half) * HID + n];
        b.y = Wn2[(k + 2 * half + 1) * HID + n];
        acc2 = __builtin_amdgcn_wmma_f32_16x16x4_f32(false, a, false, b,
                                                     (short)0, acc2, false, false);
    }

    // ---- store: VGPR r -> row m0 + r + 8*half, 16 contiguous cols per lane-half (64B chunks) ----
    #pragma unroll
    for (int r = 0; r < 8; ++r) {
        int m = r + 8 * half;
        out[(size_t)(m0 + m) * HID + n] = acc2[r];
    }
}

extern "C" void kernel_launch(void* const* d_in, const int* in_sizes, int n_in,
                              void* d_out, int out_size, void* d_ws, size_t ws_size,
                              hipStream_t stream) {
    // setup_inputs order:
    // 0:h 1:edges 2:distances 3:W_e1 4:b_e1 5:W_e2 6:b_e2 7:W_ei 8:b_ei 9:W_n1 10:b_n1 11:W_n2 12:b_n2
    const float* h     = (const float*)d_in[0];
    const int*   edges = (const int*)d_in[1];
    const float* dist  = (const float*)d_in[2];
    const float* Wn1   = (const float*)d_in[9];
    const float* bn1   = (const float*)d_in[10];
    const float* Wn2   = (const float*)d_in[11];
    const float* bn2   = (const float*)d_in[12];
    float*       out   = (float*)d_out;
    float*       agg   = (float*)d_ws;            // 50000 floats = 200KB scratch

    GCL_zero_agg<<<(N_NODES + 255) / 256, 256, 0, stream>>>(agg);
    GCL_scatter<<<(N_EDGES + 255) / 256, 256, 0, stream>>>(edges, dist, agg);
    GCL_node_mlp<<<N_NODES / 16, 256, 0, stream>>>(h, agg, Wn1, bn1, Wn2, bn2, out);
}